// Shapelet_37529424233204
// MI455X (gfx1250) — compile-verified
//
#include <hip/hip_runtime.h>
#include <stdint.h>

// Problem constants (from reference): B=16, M=8, T=1024, N=32, L=32, STRIDE=1
// t = 993 sliding windows; outputs: max_p (16x256) then dmin (16x256), fp32.

#define T_WINDOWS 993

__device__ __forceinline__ void async_b128(uint32_t lds_byte_addr, uint32_t voff_bytes,
                                           const void* sbase) {
    // GLOBAL_LOAD_ASYNC_TO_LDS_B128, GVS mode: mem = SADDR + VADDR, LDS addr from vdst.
    // Tracked by ASYNCcnt (CDNA5 async-to-LDS path).
    asm volatile("global_load_async_to_lds_b128 %0, %1, %2"
                 :: "v"(lds_byte_addr), "v"(voff_bytes), "s"(sbase)
                 : "memory");
}

__global__ __launch_bounds__(256) void shapelet_kernel(const float* __restrict__ x,
                                                       const float* __restrict__ w,
                                                       float* __restrict__ out) {
    __shared__ float x_s[1024];   // one (b,m) row of x
    __shared__ float w_s[1024];   // w[:, m, :]  (32 rows x 32 floats, 128B rows)
    __shared__ float dmin_s[32];  // per-n running min (raw sum, scaled at the end)

    const int tid = threadIdx.x;
    const int bid = blockIdx.x;          // bid = b*8 + m
    const int m   = bid & 7;

    const uint32_t xs = (uint32_t)(uintptr_t)x_s;   // low 32 bits of flat addr = LDS byte offset
    const uint32_t ws = (uint32_t)(uintptr_t)w_s;

    // ---- async stage: x row (4KB) and weight slice (4KB) into LDS ----
    const float* xrow  = x + (size_t)bid * 1024;    // (b*M + m) * T
    const float* wbase = w + m * 32;                // row n lives at +n*256 floats
    async_b128(xs + tid * 16u, tid * 16u, xrow);
    {
        const uint32_t n = (uint32_t)tid >> 3, c = (uint32_t)tid & 7;
        async_b128(ws + tid * 16u, n * 1024u + c * 16u, wbase);  // bytes: n*(M*L*4) + c*16
    }
    if (tid < 32) dmin_s[tid] = __builtin_inff();

    asm volatile("s_wait_asynccnt 0" ::: "memory");
    __syncthreads();

    const uint32_t dmin_base = (uint32_t)(uintptr_t)dmin_s;

    // ---- core: two passes, each handles windows tA (<=511) and tB = tA+512 ----
    for (int jj = 0; jj < 2; ++jj) {
        const int  tA = tid + jj * 256;          // always valid (<= 511 <= 992)
        int        tB = tA + 512;                // up to 1023
        const bool vB = (tB < T_WINDOWS);
        if (!vB) tB = 0;                         // keep LDS reads in-bounds; result discarded

        float xa[32], xb[32];
        #pragma unroll
        for (int l = 0; l < 32; ++l) { xa[l] = x_s[tA + l]; xb[l] = x_s[tB + l]; }

        for (int n = 0; n < 32; ++n) {
            const float4* wr = (const float4*)&w_s[n * 32];   // 16B-aligned broadcast reads
            float accA = 0.0f, accB = 0.0f;
            #pragma unroll
            for (int q = 0; q < 8; ++q) {
                const float4 wv = wr[q];
                accA += __builtin_fabsf(xa[4*q+0] - wv.x); accB += __builtin_fabsf(xb[4*q+0] - wv.x);
                accA += __builtin_fabsf(xa[4*q+1] - wv.y); accB += __builtin_fabsf(xb[4*q+1] - wv.y);
                accA += __builtin_fabsf(xa[4*q+2] - wv.z); accB += __builtin_fabsf(xb[4*q+2] - wv.z);
                accA += __builtin_fabsf(xa[4*q+3] - wv.w); accB += __builtin_fabsf(xb[4*q+3] - wv.w);
            }
            float v = fminf(accA, vB ? accB : __builtin_inff());

            // wave32 min-reduction: each lane is a distinct t
            #pragma unroll
            for (int s = 16; s > 0; s >>= 1) v = fminf(v, __shfl_xor(v, s, 32));

            if ((tid & 31) == 0) {
                asm volatile("ds_min_num_f32 %0, %1"
                             :: "v"(dmin_base + (uint32_t)(n * 4)), "v"(v)
                             : "memory");
            }
        }
    }

    asm volatile("s_wait_dscnt 0" ::: "memory");
    __syncthreads();

    // ---- emit: dmin and max_p = exp(-dmin^2) ----
    if (tid < 32) {
        const float d = dmin_s[tid] * (1.0f / 32.0f);   // mean over L
        const float p = expf(-d * d);
        const int   o = (bid >> 3) * 256 + tid * 8 + (bid & 7);  // b*256 + n*8 + m
        out[o]        = p;   // max_p
        out[4096 + o] = d;   // dmin
    }
}

extern "C" void kernel_launch(void* const* d_in, const int* in_sizes, int n_in,
                              void* d_out, int out_size, void* d_ws, size_t ws_size,
                              hipStream_t stream) {
    const float* x = (const float*)d_in[0];   // (16, 8, 1024) fp32
    const float* w = (const float*)d_in[1];   // (32, 8, 32)  fp32
    float* out = (float*)d_out;               // 8192 fp32: max_p then dmin
    hipLaunchKernelGGL(shapelet_kernel, dim3(128), dim3(256), 0, stream, x, w, out);
}